// GraphConv_43662637531370
// MI455X (gfx1250) — compile-verified
//
#include <hip/hip_runtime.h>

// out = A @ x, A sparse COO: out[row[e], :] += edge_vals[e] * x[col[e], :]
// N = 500K nodes, E = 8M edges, D = 64 features (f32).
//
// Memory-bound (~350 MB min HBM traffic -> ~15us floor @ 23.3 TB/s).
// x (128 MB) is L2-resident on MI455X (192 MB L2); atomics resolve in L2.
// Strategy per wave32: own 32 edges, stage (row,col,val) triples into LDS
// with gfx1250 async-to-LDS copies (double buffered, ASYNCcnt, marked
// non-temporal so the stream-once edge data doesn't evict the reused x/out
// working set from L2), then broadcast each edge via v_readlane and have
// 32 lanes cover the 64 features as float2 loads + 2x global_atomic_add_f32.

#define WAVES_PER_BLOCK 8
#define THREADS (WAVES_PER_BLOCK * 32)
#define GRID_BLOCKS 4096

__global__ __launch_bounds__(256) void zero_f32(float* __restrict__ p, int n) {
    int i = (blockIdx.x * blockDim.x + threadIdx.x) * 4;
    if (i + 3 < n) {
        *reinterpret_cast<float4*>(p + i) = float4{0.f, 0.f, 0.f, 0.f};
    } else {
        for (int k = i; k < n; ++k) p[k] = 0.f;
    }
}

__global__ __launch_bounds__(THREADS) void spmm_coo_atomic(
    const float* __restrict__ x,        // [N, 64]
    const float* __restrict__ edge_vals,// [E]
    const int*   __restrict__ row,      // [E]
    const int*   __restrict__ col,      // [E]
    float*       __restrict__ out,      // [N, 64]
    int n_edges)
{
    // Per-wave double-buffered edge staging: [row x32][col x32][val x32]
    __shared__ int ebuf[WAVES_PER_BLOCK][2][96];

    const int lane   = threadIdx.x & 31;
    const int wib    = threadIdx.x >> 5;
    const int gwid   = (blockIdx.x * THREADS + threadIdx.x) >> 5;
    const int nwaves = (GRID_BLOCKS * THREADS) >> 5;
    const int nbatch = (n_edges + 31) >> 5;

    if (gwid >= nbatch) return;

    // LDS byte addresses for the two buffers (low 32 bits of flat shared
    // address == LDS offset per the gfx1250 aperture mapping).
    const unsigned lds0 =
        (unsigned)(unsigned long long)(const void*)&ebuf[wib][0][0];
    const unsigned lds1 =
        (unsigned)(unsigned long long)(const void*)&ebuf[wib][1][0];

    // Stage one 32-edge batch into an LDS buffer with async copies.
    // Each lane moves one b32 of row, col, val (coalesced stream). The edge
    // arrays are touched exactly once -> non-temporal so they don't evict
    // the L2-resident x / out working set.
    auto stage = [&](int b, unsigned ldsbase) {
        int e = b * 32 + lane;
        if (e >= n_edges) e = n_edges - 1;        // tail clamp (unused lanes)
        const unsigned eb = (unsigned)e * 4u;     // byte offset into edge arrays
        const unsigned d0 = ldsbase + (unsigned)lane * 4u;
        const unsigned d1 = d0 + 128u;
        const unsigned d2 = d0 + 256u;
        asm volatile("global_load_async_to_lds_b32 %0, %1, %2 th:TH_LOAD_NT"
                     :: "v"(d0), "v"(eb), "s"(row) : "memory");
        asm volatile("global_load_async_to_lds_b32 %0, %1, %2 th:TH_LOAD_NT"
                     :: "v"(d1), "v"(eb), "s"(col) : "memory");
        asm volatile("global_load_async_to_lds_b32 %0, %1, %2 th:TH_LOAD_NT"
                     :: "v"(d2), "v"(eb), "s"(edge_vals) : "memory");
    };

    // Consume a staged batch: broadcast each edge across the wave, 32 lanes
    // cover 64 features as float2, scatter with no-return f32 atomics.
    auto process = [&](int b, int sel) {
        const int   r = ebuf[wib][sel][lane];
        const int   c = ebuf[wib][sel][32 + lane];
        const float v = __int_as_float(ebuf[wib][sel][64 + lane]);

        // Warm L2 with this lane's own source row (2 cachelines, 256 B).
        const float* xpf = x + (size_t)c * 64;
        __builtin_prefetch(xpf, 0, 3);
        __builtin_prefetch(xpf + 32, 0, 3);

        int cnt = n_edges - b * 32;
        if (cnt > 32) cnt = 32;
        const unsigned f = (unsigned)lane * 2u;

        for (int j = 0; j < cnt; ++j) {
            const int   cj = __builtin_amdgcn_readlane(c, j);
            const int   rj = __builtin_amdgcn_readlane(r, j);
            const float vj =
                __int_as_float(__builtin_amdgcn_readlane(__float_as_int(v), j));

            const float2 xv =
                *reinterpret_cast<const float2*>(x + (size_t)cj * 64 + f);
            float* o = out + (size_t)rj * 64 + f;
            unsafeAtomicAdd(o,     xv.x * vj);   // global_atomic_add_f32 (no rtn)
            unsafeAtomicAdd(o + 1, xv.y * vj);
        }
    };

    int b  = gwid;
    int bn = b + nwaves;
    int sel = 0;
    stage(b, lds0);
    for (;;) {
        if (bn < nbatch) {
            stage(bn, sel ? lds0 : lds1);
            // 6 outstanding async loads; wait until only next-batch's 3 remain.
            asm volatile("s_wait_asynccnt 0x3" ::: "memory");
        } else {
            asm volatile("s_wait_asynccnt 0x0" ::: "memory");
        }
        process(b, sel);
        if (bn >= nbatch) break;
        b = bn;
        bn += nwaves;
        sel ^= 1;
    }
}

extern "C" void kernel_launch(void* const* d_in, const int* in_sizes, int n_in,
                              void* d_out, int out_size, void* d_ws, size_t ws_size,
                              hipStream_t stream) {
    const float* x         = (const float*)d_in[0];
    const float* edge_vals = (const float*)d_in[1];
    const int*   row       = (const int*)d_in[2];
    const int*   col       = (const int*)d_in[3];
    float*       out       = (float*)d_out;
    const int n_edges = in_sizes[1];

    // Zero the accumulator every call (deterministic; harness poisons once).
    const int zthreads = 256;
    const int zblocks  = (out_size / 4 + zthreads - 1) / zthreads + 1;
    zero_f32<<<zblocks, zthreads, 0, stream>>>(out, out_size);

    spmm_coo_atomic<<<GRID_BLOCKS, THREADS, 0, stream>>>(
        x, edge_vals, row, col, out, n_edges);
}